// action_network_50749333569733
// MI455X (gfx1250) — compile-verified
//
#include <hip/hip_runtime.h>
#include <math.h>

typedef __attribute__((ext_vector_type(2))) float v2f;
typedef __attribute__((ext_vector_type(8))) float v8f;

#define N_NODES 100000
#define N_EDGES 20000
#define N_PAIRS 3200000
#define DDIM    512

__device__ __forceinline__ float gelu_exact(float v) {
    return 0.5f * v * (1.0f + erff(v * 0.70710678118654752f));
}

// ---------------------------------------------------------------------------
// Kernel 1: fused dual GEMM via V_WMMA_F32_16X16X4_F32.
//   m_ji = gelu(x @ Wmsg^T)   (cols 0-3 of the 16-wide packed B)
//   u    =       x @ Wupd^T   (cols 4-7)
// One wave handles one 16-row tile of x; K=512 -> 128 chained WMMAs.
// B is staged in LDS as Wpack[d][16] (stride 16 floats => conflict-free:
// lanes 0-15 read row k+r, lanes 16-31 read row k+r+2 -> disjoint bank sets).
// A layout (ISA 7.12.2, 32-bit A 16x4): lane L (L<16) row M=L holds K=0,1;
// lanes 16-31 hold K=2,3.  B (4x16) analogous: VGPR r = row K=r (+2 for the
// upper half-lanes).  D layout: reg r -> row r + 8*(lane>=16), col = lane&15.
// ---------------------------------------------------------------------------
__global__ __launch_bounds__(256) void k_gemm(const float* __restrict__ x,
                                              const float* __restrict__ Wmsg,
                                              const float* __restrict__ Wupd,
                                              float* __restrict__ m_ji,
                                              float* __restrict__ u) {
    __shared__ float Wpack[DDIM * 16];     // [d][k]: k 0-3 = Wmsg, 4-7 = Wupd, 8-15 = 0
    const int tid = threadIdx.x;
    for (int i = tid; i < DDIM * 16; i += 256) {
        const int d = i >> 4, k = i & 15;
        float w = 0.0f;
        if (k < 4)      w = Wmsg[k * DDIM + d];
        else if (k < 8) w = Wupd[(k - 4) * DDIM + d];
        Wpack[i] = w;
    }
    __syncthreads();

    const int wave    = blockIdx.x * (blockDim.x >> 5) + (tid >> 5);
    const int n_tiles = N_NODES / 16;      // 6250, exact
    if (wave >= n_tiles) return;           // wave-uniform: EXEC stays all-1s

    const int lane = tid & 31;
    const int half = lane >> 4;            // 0: K={0,1}, 1: K={2,3}
    const int l15  = lane & 15;
    const int roff = half * 2;
    const int row  = wave * 16 + l15;

    const float* xrow = x + (size_t)row * DDIM + roff;   // 8B aligned
    const float* bcol = &Wpack[roff * 16 + l15];

    v8f acc = {};
    #pragma unroll 8
    for (int k = 0; k < DDIM; k += 4) {
        const float2 av = *reinterpret_cast<const float2*>(xrow + k);
        v2f a, b;
        a.x = av.x;                        // K = k + roff
        a.y = av.y;                        // K = k + roff + 1
        b.x = bcol[k * 16];                // B row K = k + roff,   col l15
        b.y = bcol[k * 16 + 16];           // B row K = k + roff+1, col l15
        acc = __builtin_amdgcn_wmma_f32_16x16x4_f32(
            /*neg_a=*/false, a, /*neg_b=*/false, b,
            /*c_mod=*/(short)0, acc, /*reuse_a=*/false, /*reuse_b=*/false);
    }

    if (l15 < 8) {                         // only cols 0-7 carry data
        const int col = l15;
        #pragma unroll
        for (int r = 0; r < 8; ++r) {
            const int orow = wave * 16 + r + 8 * half;
            const float v  = acc[r];
            if (col < 4) m_ji[orow * 4 + col]     = gelu_exact(v);
            else         u[orow * 4 + (col - 4)]  = v;
        }
    }
}

// ---------------------------------------------------------------------------
// Kernel 2: pairs -> edge accumulation.  m_ji is 1.6MB (lives in L2).
// ---------------------------------------------------------------------------
__global__ __launch_bounds__(256) void k_scatter_edge(const float* __restrict__ m_ji,
                                                      const int* __restrict__ pair_node,
                                                      const int* __restrict__ pair_edge,
                                                      float* __restrict__ e_sum,
                                                      float* __restrict__ e_cnt) {
    const int p = blockIdx.x * blockDim.x + threadIdx.x;
    if (p >= N_PAIRS) return;
    const int n = pair_node[p];
    const int e = pair_edge[p];
    const float4 m = *reinterpret_cast<const float4*>(m_ji + (size_t)n * 4);
    atomicAdd(&e_sum[e * 4 + 0], m.x);
    atomicAdd(&e_sum[e * 4 + 1], m.y);
    atomicAdd(&e_sum[e * 4 + 2], m.z);
    atomicAdd(&e_sum[e * 4 + 3], m.w);
    atomicAdd(&e_cnt[e], 1.0f);
}

// Kernel 3: edge mean (in place: e_sum becomes e_feat).
__global__ __launch_bounds__(256) void k_edge_mean(float* __restrict__ e_sum,
                                                   const float* __restrict__ e_cnt) {
    const int e = blockIdx.x * blockDim.x + threadIdx.x;
    if (e >= N_EDGES) return;
    const float inv = 1.0f / fmaxf(e_cnt[e], 1.0f);
    e_sum[e * 4 + 0] *= inv;
    e_sum[e * 4 + 1] *= inv;
    e_sum[e * 4 + 2] *= inv;
    e_sum[e * 4 + 3] *= inv;
}

// Kernel 4: pairs -> node accumulation (e_feat is 320KB, lives in L2).
__global__ __launch_bounds__(256) void k_scatter_node(const float* __restrict__ e_feat,
                                                      const int* __restrict__ pair_node,
                                                      const int* __restrict__ pair_edge,
                                                      float* __restrict__ m_sum,
                                                      float* __restrict__ m_cnt) {
    const int p = blockIdx.x * blockDim.x + threadIdx.x;
    if (p >= N_PAIRS) return;
    const int n = pair_node[p];
    const int e = pair_edge[p];
    const float4 f = *reinterpret_cast<const float4*>(e_feat + (size_t)e * 4);
    atomicAdd(&m_sum[n * 4 + 0], f.x);
    atomicAdd(&m_sum[n * 4 + 1], f.y);
    atomicAdd(&m_sum[n * 4 + 2], f.z);
    atomicAdd(&m_sum[n * 4 + 3], f.w);
    atomicAdd(&m_cnt[n], 1.0f);
}

// Kernel 5: h = gelu(u + b + m_i), then log_softmax over the 4 classes.
__global__ __launch_bounds__(256) void k_final(const float* __restrict__ u,
                                               const float* __restrict__ m_sum,
                                               const float* __restrict__ m_cnt,
                                               const float* __restrict__ b_upd,
                                               float* __restrict__ out) {
    const int n = blockIdx.x * blockDim.x + threadIdx.x;
    if (n >= N_NODES) return;
    const float inv = 1.0f / fmaxf(m_cnt[n], 1.0f);
    const float4 uv = *reinterpret_cast<const float4*>(u + (size_t)n * 4);
    const float4 mv = *reinterpret_cast<const float4*>(m_sum + (size_t)n * 4);
    float h[4];
    h[0] = gelu_exact(uv.x + b_upd[0] + mv.x * inv);
    h[1] = gelu_exact(uv.y + b_upd[1] + mv.y * inv);
    h[2] = gelu_exact(uv.z + b_upd[2] + mv.z * inv);
    h[3] = gelu_exact(uv.w + b_upd[3] + mv.w * inv);
    const float mx = fmaxf(fmaxf(h[0], h[1]), fmaxf(h[2], h[3]));
    const float s  = expf(h[0] - mx) + expf(h[1] - mx) + expf(h[2] - mx) + expf(h[3] - mx);
    const float lse = mx + logf(s);
    out[n * 4 + 0] = h[0] - lse;
    out[n * 4 + 1] = h[1] - lse;
    out[n * 4 + 2] = h[2] - lse;
    out[n * 4 + 3] = h[3] - lse;
}

// Zero the atomic accumulators every call (graph replays don't re-poison).
__global__ __launch_bounds__(256) void k_zero(float* __restrict__ p, int n) {
    const int i = blockIdx.x * blockDim.x + threadIdx.x;
    if (i < n) p[i] = 0.0f;
}

extern "C" void kernel_launch(void* const* d_in, const int* in_sizes, int n_in,
                              void* d_out, int out_size, void* d_ws, size_t ws_size,
                              hipStream_t stream) {
    (void)in_sizes; (void)n_in; (void)out_size; (void)ws_size;
    const float* x         = (const float*)d_in[0];
    const int*   pair_node = (const int*)  d_in[1];
    const int*   pair_edge = (const int*)  d_in[2];
    const float* Wmsg      = (const float*)d_in[3];
    const float* Wupd      = (const float*)d_in[4];
    const float* b_upd     = (const float*)d_in[5];
    float* out = (float*)d_out;

    // Workspace layout (floats), all 16B aligned:
    float* ws    = (float*)d_ws;
    float* m_ji  = ws;                 // N*4 = 400000
    float* u     = ws + 400000;        // N*4 = 400000
    float* e_sum = ws + 800000;        // E*4 =  80000
    float* e_cnt = ws + 880000;        // E   =  20000
    float* m_sum = ws + 900000;        // N*4 = 400000
    float* m_cnt = ws + 1300000;       // N   = 100000   (total 1.4M floats = 5.6MB)

    const int ZN = 600000;             // e_sum..m_cnt contiguous region
    k_zero<<<(ZN + 255) / 256, 256, 0, stream>>>(e_sum, ZN);

    const int n_tiles = N_NODES / 16;                  // 6250
    k_gemm<<<(n_tiles + 7) / 8, 256, 0, stream>>>(x, Wmsg, Wupd, m_ji, u);

    k_scatter_edge<<<(N_PAIRS + 255) / 256, 256, 0, stream>>>(m_ji, pair_node, pair_edge, e_sum, e_cnt);
    k_edge_mean<<<(N_EDGES + 255) / 256, 256, 0, stream>>>(e_sum, e_cnt);
    k_scatter_node<<<(N_PAIRS + 255) / 256, 256, 0, stream>>>(e_sum, pair_node, pair_edge, m_sum, m_cnt);
    k_final<<<(N_NODES + 255) / 256, 256, 0, stream>>>(u, m_sum, m_cnt, b_upd, out);
}